// GalleryFilterNetwork_43293270344291
// MI455X (gfx1250) — compile-verified
//
#include <hip/hip_runtime.h>
#include <math.h>

// Problem constants (from reference): Q=256 queries, I=2048 gallery, D=256 dims.
#define QN 256
#define IN 2048
#define DN 256
#define SE_INV_TEMP 5.0f      // 1/0.2
#define BN_EPS 1e-5f

typedef __attribute__((ext_vector_type(2))) float v2f;
typedef __attribute__((ext_vector_type(8))) float v8f;

// ---- block-wide sum reduction over 256 threads (8 waves, wave32) ----
__device__ inline float block_reduce_sum(float v, float* sh) {
    const int t = threadIdx.x;
    sh[t] = v;
    __syncthreads();
    #pragma unroll
    for (int off = 128; off >= 1; off >>= 1) {
        if (t < off) sh[t] += sh[t + off];
        __syncthreads();
    }
    float r = sh[0];
    __syncthreads();
    return r;
}

// ---------------------------------------------------------------------------
// Kernel 1: per-query precompute.
//   gate = sigmoid(l2n(qf)/T);  u = l2n(bn(l2n(qimg)*gate))
//   A = u*gate*s ; P = (gate*s)^2 ; R = gate*s*b ; c = dot(u,b) ; sumb2 = dot(b,b)
// one block per q, one thread per d.
// ---------------------------------------------------------------------------
__global__ __launch_bounds__(DN) void gfn_query_prep(
    const float* __restrict__ qf, const float* __restrict__ qimg,
    const float* __restrict__ gamma, const float* __restrict__ beta,
    const float* __restrict__ mean, const float* __restrict__ var,
    float* __restrict__ A, float* __restrict__ P, float* __restrict__ Rm,
    float* __restrict__ cvec, float* __restrict__ sumb2) {
    __shared__ float sh[DN];
    const int q = blockIdx.x, d = threadIdx.x;

    float x  = qf[q * DN + d];
    float nx = sqrtf(block_reduce_sum(x * x, sh));
    float xq = x / fmaxf(nx, 1e-12f);
    float gate = 1.0f / (1.0f + expf(-xq * SE_INV_TEMP));

    float y  = qimg[q * DN + d];
    float ny = sqrtf(block_reduce_sum(y * y, sh));
    float yq = y / fmaxf(ny, 1e-12f);

    float s  = gamma[d] * rsqrtf(var[d] + BN_EPS);
    float bb = beta[d] - mean[d] * s;

    float fq = yq * gate * s + bb;               // bn(yq*gate)
    float nf = sqrtf(block_reduce_sum(fq * fq, sh));
    float u  = fq / fmaxf(nf, 1e-12f);           // l2n(fused_query)

    float gs = gate * s;
    A [q * DN + d] = u * gs;
    P [q * DN + d] = gs * gs;
    Rm[q * DN + d] = gs * bb;

    float c = block_reduce_sum(u * bb, sh);
    if (d == 0) cvec[q] = c;

    if (q == 0) {                                // uniform per-block branch: safe
        float sb = block_reduce_sum(bb * bb, sh);
        if (d == 0) *sumb2 = sb;
    }
}

// ---------------------------------------------------------------------------
// Kernel 2: gallery precompute: YG = l2n(gal) ; YG2 = YG^2
// ---------------------------------------------------------------------------
__global__ __launch_bounds__(DN) void gfn_gal_prep(
    const float* __restrict__ gal, float* __restrict__ YG, float* __restrict__ YG2) {
    __shared__ float sh[DN];
    const int i = blockIdx.x, d = threadIdx.x;
    float v = gal[i * DN + d];
    float n = sqrtf(block_reduce_sum(v * v, sh));
    float vn = v / fmaxf(n, 1e-12f);
    YG [i * DN + d] = vn;
    YG2[i * DN + d] = vn * vn;
}

// ---------------------------------------------------------------------------
// Kernel 3: triple GEMM via V_WMMA_F32_16X16X4_F32 + fused sigmoid epilogue.
//   scores[q,i] = sigmoid( (A@yg^T + c) / max(sqrt(P@yg2^T + 2*R@yg^T + sumb2), 1e-12) )
// 8 waves/block, each wave computes one 16(q) x 16(i) tile; 256 blocks -> 2048 tiles.
//
// f32 WMMA A-matrix 16x4 layout (per ISA 7.12.2): lane L holds M = L%16,
// VGPR j with half = L/16 -> K = 2*half + j.  B-matrix 4x16: N = L%16, same K
// striping.  Both map to contiguous float2 loads from row-major [*, D] buffers.
// ---------------------------------------------------------------------------
__global__ __launch_bounds__(256) void gfn_score_wmma(
    const float* __restrict__ A, const float* __restrict__ P, const float* __restrict__ Rm,
    const float* __restrict__ YG, const float* __restrict__ YG2,
    const float* __restrict__ cvec, const float* __restrict__ sumb2p,
    float* __restrict__ out) {
    const int wave = threadIdx.x >> 5;
    const int lane = threadIdx.x & 31;
    const int tile = blockIdx.x * 8 + wave;      // 0..2047
    const int qt = tile >> 7;                    // 16 q-tiles
    const int it = tile & 127;                   // 128 i-tiles
    const int q0 = qt * 16, i0 = it * 16;
    const int half = lane >> 4, lm = lane & 15;

    const float* aP  = A   + (q0 + lm) * DN + 2 * half;
    const float* pP  = P   + (q0 + lm) * DN + 2 * half;
    const float* rP  = Rm  + (q0 + lm) * DN + 2 * half;
    const float* gP  = YG  + (i0 + lm) * DN + 2 * half;
    const float* g2P = YG2 + (i0 + lm) * DN + 2 * half;

    v8f accA = {}, accP = {}, accR = {};
    #pragma unroll 4
    for (int k = 0; k < DN; k += 4) {
        v2f a  = *(const v2f*)(aP  + k);
        v2f p  = *(const v2f*)(pP  + k);
        v2f r  = *(const v2f*)(rP  + k);
        v2f g  = *(const v2f*)(gP  + k);
        v2f g2 = *(const v2f*)(g2P + k);
        accA = __builtin_amdgcn_wmma_f32_16x16x4_f32(false, a, false, g,  (short)0, accA, false, false);
        accP = __builtin_amdgcn_wmma_f32_16x16x4_f32(false, p, false, g2, (short)0, accP, false, false);
        accR = __builtin_amdgcn_wmma_f32_16x16x4_f32(false, r, false, g,  (short)0, accR, false, false);
    }

    const float sumb2 = *sumb2p;
    #pragma unroll
    for (int rr = 0; rr < 8; ++rr) {
        const int q = q0 + rr + 8 * half;        // C/D layout: M = vgpr + 8*(lane/16)
        const int i = i0 + lm;                   //             N = lane%16
        float num = accA[rr] + cvec[q];
        float den2 = accP[rr] + 2.0f * accR[rr] + sumb2;
        float den = fmaxf(sqrtf(fmaxf(den2, 0.0f)), 1e-12f);
        float xv = num / den;
        out[q * IN + i] = 1.0f / (1.0f + expf(-xv));
    }
}

// ---------------------------------------------------------------------------
extern "C" void kernel_launch(void* const* d_in, const int* in_sizes, int n_in,
                              void* d_out, int out_size, void* d_ws, size_t ws_size,
                              hipStream_t stream) {
    (void)in_sizes; (void)n_in; (void)out_size; (void)ws_size;
    const float* qf    = (const float*)d_in[0];  // [Q,D]
    const float* qimg  = (const float*)d_in[1];  // [Q,D]
    const float* gal   = (const float*)d_in[2];  // [I,D]
    const float* gamma = (const float*)d_in[3];  // [D]
    const float* beta  = (const float*)d_in[4];  // [D]
    const float* mean  = (const float*)d_in[5];  // [D]
    const float* var   = (const float*)d_in[6];  // [D]
    float* out = (float*)d_out;                  // [Q,I]

    float* ws = (float*)d_ws;
    float* A     = ws;                                  // Q*D
    float* P     = A + QN * DN;                         // Q*D
    float* Rm    = P + QN * DN;                         // Q*D
    float* YG    = Rm + QN * DN;                        // I*D
    float* YG2   = YG + IN * DN;                        // I*D
    float* cvec  = YG2 + IN * DN;                       // Q
    float* sumb2 = cvec + QN;                           // 1

    gfn_query_prep<<<QN, DN, 0, stream>>>(qf, qimg, gamma, beta, mean, var,
                                          A, P, Rm, cvec, sumb2);
    gfn_gal_prep<<<IN, DN, 0, stream>>>(gal, YG, YG2);
    gfn_score_wmma<<<(QN / 16) * (IN / 16) / 8, 256, 0, stream>>>(
        A, P, Rm, YG, YG2, cvec, sumb2, out);
}